// NeuralODEPKModel_71932112273720
// MI455X (gfx1250) — compile-verified
//
#include <hip/hip_runtime.h>
#include <math.h>

// ---------------------------------------------------------------------------
// Neural ODE (encoder -> RK4 x99 -> decoder@every step) on gfx1250.
// One wave32 owns a 16-sample batch tile; state kept transposed (latent x batch)
// so all GEMMs are V_WMMA_F32_16X16X4_F32 with constant A (= W^T) operands.
// Layout fixups between chained WMMAs are half-wave swaps via ds_swizzle.
// ---------------------------------------------------------------------------

typedef __attribute__((ext_vector_type(2))) float v2f;
typedef __attribute__((ext_vector_type(8))) float v8f;

#if __has_builtin(__builtin_amdgcn_tanhf)
#define TANH_F32(x) __builtin_amdgcn_tanhf(x)
#else
#define TANH_F32(x) tanhf(x)
#endif

__device__ __forceinline__ v8f wmma4(v2f a, v2f b, v8f c) {
  // D(16x16,f32) = A(16x4,f32) x B(4x16,f32) + C
  return __builtin_amdgcn_wmma_f32_16x16x4_f32(false, a, false, b, (short)0, c,
                                               false, false);
}

// lane L <-> lane L^16 (SWAPX16 group-of-32 swizzle; no LDS memory traffic)
__device__ __forceinline__ float swap16(float x) {
  return __int_as_float(__builtin_amdgcn_ds_swizzle(__float_as_int(x), 0x401F));
}

#define HID 50
#define LAT 16

__global__ __launch_bounds__(256) void node_rk4_kernel(
    const float* __restrict__ x0, const float* __restrict__ tv,
    const float* __restrict__ We1, const float* __restrict__ be1,
    const float* __restrict__ We2, const float* __restrict__ be2,
    const float* __restrict__ Wo1, const float* __restrict__ bo1,
    const float* __restrict__ Wo2, const float* __restrict__ bo2,
    const float* __restrict__ Wd1, const float* __restrict__ bd1,
    const float* __restrict__ Wd2, const float* __restrict__ bd2,
    float* __restrict__ out, int B, int T) {
  const int lane = threadIdx.x & 31;
  const int wid = blockIdx.x * (blockDim.x >> 5) + (threadIdx.x >> 5);
  const int base = wid << 4;  // 16 batch samples per wave
  if (base >= B) return;      // whole-wave uniform: EXEC stays all-ones for WMMA
  const int m16 = lane & 15;
  const bool hi = lane >= 16;
  const int n = base + m16;   // this lane's batch column

  // A-operand chunk loader: A(m, k) = W[k*ld + m], rows m in lanes,
  // K=4 chunk split as {lo lanes: k=kb,kb+1 ; hi lanes: k=kb+2,kb+3}.
  auto ald = [&](const float* W, int ld, int K, int M, int m, int kb) -> v2f {
    int k0 = kb + (hi ? 2 : 0);
    float a0 = (m < M && k0 < K) ? W[k0 * ld + m] : 0.f;
    float a1 = (m < M && (k0 + 1) < K) ? W[(k0 + 1) * ld + m] : 0.f;
    v2f r = {a0, a1};
    return r;
  };

  const v2f onesB = {hi ? 0.f : 1.f, 0.f};  // B chunk whose row k=kb is all-ones
  const v8f vzero = {0.f, 0.f, 0.f, 0.f, 0.f, 0.f, 0.f, 0.f};

  // ---- preload persistent weights in A layout ----
  v2f wo1A[4][4], wo1Ab[4];  // Wo1^T (hid x lat) tiles+chunks, bias-as-K chunk
  v2f wd1A[4][4], wd1Ab[4];  // Wd1^T (hid x lat)
  v8f wd2v[4];               // decoder 50->1 weights, C/D-layout aligned
  v2f wo2A[13];              // Wo2^T (lat x hid), K padded 50->52
  v8f bo2C;                  // latent bias as initial C
#pragma unroll
  for (int tt = 0; tt < 4; ++tt) {
#pragma unroll
    for (int c = 0; c < 4; ++c) {
      wo1A[tt][c] = ald(Wo1, HID, LAT, HID, 16 * tt + m16, 4 * c);
      wd1A[tt][c] = ald(Wd1, HID, LAT, HID, 16 * tt + m16, 4 * c);
    }
    int mb = 16 * tt + m16;
    {
      v2f a = {(!hi && mb < HID) ? bo1[mb] : 0.f, 0.f};
      wo1Ab[tt] = a;
      v2f b = {(!hi && mb < HID) ? bd1[mb] : 0.f, 0.f};
      wd1Ab[tt] = b;
    }
#pragma unroll
    for (int v = 0; v < 8; ++v) {
      int m = 16 * tt + v + (hi ? 8 : 0);
      wd2v[tt][v] = (m < HID) ? Wd2[m] : 0.f;
    }
  }
#pragma unroll
  for (int c = 0; c < 13; ++c) wo2A[c] = ald(Wo2, LAT, HID, LAT, m16, 4 * c);
#pragma unroll
  for (int v = 0; v < 8; ++v) bo2C[v] = bo2[v + (hi ? 8 : 0)];
  const float bd2s = bd2[0];

  // C/D layout (row m: vgpr m%8, half m>=8) -> B chunks (row k: reg k%2, half k%4>=2)
  auto convfull = [&](const v8f& cd, v2f* bc) {
    float s0 = swap16(cd[0]), s1 = swap16(cd[1]), s2 = swap16(cd[2]),
          s3 = swap16(cd[3]), s4 = swap16(cd[4]), s5 = swap16(cd[5]),
          s6 = swap16(cd[6]), s7 = swap16(cd[7]);
    v2f b0 = {hi ? s2 : cd[0], hi ? s3 : cd[1]};  // rows 0..3
    v2f b1 = {hi ? s6 : cd[4], hi ? s7 : cd[5]};  // rows 4..7
    v2f b2 = {hi ? cd[2] : s0, hi ? cd[3] : s1};  // rows 8..11
    v2f b3 = {hi ? cd[6] : s4, hi ? cd[7] : s5};  // rows 12..15
    bc[0] = b0; bc[1] = b1; bc[2] = b2; bc[3] = b3;
  };

  // f(z) = Wo2^T @ tanh(Wo1^T @ z^T + bo1) + bo2  (all transposed, B chunks in, C/D out)
  auto feval = [&](const v2f* inB) -> v8f {
    v8f g0 = bo2C, g1 = vzero;  // dual accumulators for WMMA ILP
#pragma unroll
    for (int tt = 0; tt < 4; ++tt) {
      v8f h = wmma4(wo1Ab[tt], onesB, vzero);  // bias via extra-K chunk
#pragma unroll
      for (int c = 0; c < 4; ++c) h = wmma4(wo1A[tt][c], inB[c], h);
      v8f th;
#pragma unroll
      for (int v = 0; v < 8; ++v) th[v] = TANH_F32(h[v]);
      if (tt < 3) {
        v2f bc[4];
        convfull(th, bc);
#pragma unroll
        for (int c = 0; c < 4; ++c) {
          if (((4 * tt + c) & 1) == 0) g0 = wmma4(wo2A[4 * tt + c], bc[c], g0);
          else                         g1 = wmma4(wo2A[4 * tt + c], bc[c], g1);
        }
      } else {  // hid rows 48..51 only (52..63 are zero pad)
        v2f bc0 = {hi ? swap16(th[2]) : th[0], hi ? swap16(th[3]) : th[1]};
        g0 = wmma4(wo2A[12], bc0, g0);
      }
    }
#pragma unroll
    for (int v = 0; v < 8; ++v) g0[v] += g1[v];
    return g0;
  };

  // decoder: y = relu(Wd1^T @ z^T + bd1) dot Wd2 + bd2, stored at timestep ts
  auto decode = [&](const v2f* zB, int ts) {
    float p = 0.f;
#pragma unroll
    for (int tt = 0; tt < 4; ++tt) {
      v8f h = wmma4(wd1Ab[tt], onesB, vzero);
#pragma unroll
      for (int c = 0; c < 4; ++c) h = wmma4(wd1A[tt][c], zB[c], h);
#pragma unroll
      for (int v = 0; v < 8; ++v)
        p = __builtin_fmaf(wd2v[tt][v], fmaxf(h[v], 0.f), p);
    }
    float tot = p + swap16(p) + bd2s;  // combine the two half-wave partials
    if (!hi && n < B) out[(size_t)ts * B + n] = tot;
  };

  // ---- encoder: z0^T = We2^T @ relu(We1^T @ x0^T + be1) + be2 ----
  v8f z;
  {
    bool ok = (!hi) && (n < B);
    v2f xB = {ok ? x0[(size_t)n * 2 + 0] : 0.f,
              ok ? x0[(size_t)n * 2 + 1] : 0.f};  // K=2 padded to 4
    v8f zc;
#pragma unroll
    for (int v = 0; v < 8; ++v) zc[v] = be2[v + (hi ? 8 : 0)];
#pragma unroll
    for (int tt = 0; tt < 4; ++tt) {
      v2f a = ald(We1, HID, 2, HID, 16 * tt + m16, 0);
      v8f c0;
#pragma unroll
      for (int v = 0; v < 8; ++v) {
        int m = 16 * tt + v + (hi ? 8 : 0);
        c0[v] = (m < HID) ? be1[m] : 0.f;
      }
      v8f h = wmma4(a, xB, c0);
#pragma unroll
      for (int v = 0; v < 8; ++v) h[v] = fmaxf(h[v], 0.f);
      if (tt < 3) {
        v2f bc[4];
        convfull(h, bc);
#pragma unroll
        for (int c = 0; c < 4; ++c)
          zc = wmma4(ald(We2, LAT, HID, LAT, m16, 4 * (4 * tt + c)), bc[c], zc);
      } else {
        v2f bc0 = {hi ? swap16(h[2]) : h[0], hi ? swap16(h[3]) : h[1]};
        zc = wmma4(ald(We2, LAT, HID, LAT, m16, 48), bc0, zc);
      }
    }
    z = zc;
  }

  v2f zB[4];
  convfull(z, zB);
  decode(zB, 0);

  // ---- RK4 time stepping ----
  for (int s = 0; s + 1 < T; ++s) {
    float dt = tv[s + 1] - tv[s];
    v8f k = feval(zB);  // k1 (reuses zB from decode)
    v8f acc = k;
    v8f zi;
    v2f ziB[4];
#pragma unroll
    for (int v = 0; v < 8; ++v) zi[v] = __builtin_fmaf(0.5f * dt, k[v], z[v]);
    convfull(zi, ziB);
    k = feval(ziB);  // k2
#pragma unroll
    for (int v = 0; v < 8; ++v) {
      acc[v] = __builtin_fmaf(2.f, k[v], acc[v]);
      zi[v] = __builtin_fmaf(0.5f * dt, k[v], z[v]);
    }
    convfull(zi, ziB);
    k = feval(ziB);  // k3
#pragma unroll
    for (int v = 0; v < 8; ++v) {
      acc[v] = __builtin_fmaf(2.f, k[v], acc[v]);
      zi[v] = __builtin_fmaf(dt, k[v], z[v]);
    }
    convfull(zi, ziB);
    k = feval(ziB);  // k4
    float w = dt * (1.f / 6.f);
#pragma unroll
    for (int v = 0; v < 8; ++v) z[v] = __builtin_fmaf(w, acc[v] + k[v], z[v]);
    convfull(z, zB);
    decode(zB, s + 1);
  }
}

extern "C" void kernel_launch(void* const* d_in, const int* in_sizes, int n_in,
                              void* d_out, int out_size, void* d_ws,
                              size_t ws_size, hipStream_t stream) {
  const float* x0 = (const float*)d_in[0];
  const float* tv = (const float*)d_in[1];
  const float* We1 = (const float*)d_in[2];
  const float* be1 = (const float*)d_in[3];
  const float* We2 = (const float*)d_in[4];
  const float* be2 = (const float*)d_in[5];
  const float* Wo1 = (const float*)d_in[6];
  const float* bo1 = (const float*)d_in[7];
  const float* Wo2 = (const float*)d_in[8];
  const float* bo2 = (const float*)d_in[9];
  const float* Wd1 = (const float*)d_in[10];
  const float* bd1 = (const float*)d_in[11];
  const float* Wd2 = (const float*)d_in[12];
  const float* bd2 = (const float*)d_in[13];
  float* out = (float*)d_out;

  int B = in_sizes[0] / 2;  // x0 is (B, 2)
  int T = in_sizes[1];
  int tiles = (B + 15) / 16;
  int wavesPerBlock = 8;  // 256 threads
  int blocks = (tiles + wavesPerBlock - 1) / wavesPerBlock;

  node_rk4_kernel<<<blocks, wavesPerBlock * 32, 0, stream>>>(
      x0, tv, We1, be1, We2, be2, Wo1, bo1, Wo2, bo2, Wd1, bd1, Wd2, bd2, out,
      B, T);
}